// YOLOLayoutLoss_77988016161165
// MI455X (gfx1250) — compile-verified
//
#include <hip/hip_runtime.h>
#include <math.h>

// ---------------- problem constants (match reference) ----------------
#define REG_MAX     16
#define NUM_CLASSES 20
#define CPRED       (4 * REG_MAX + NUM_CLASSES)   // 84
#define BATCH       32
#define NGT         200
#define NPAIR       (BATCH * NGT)                 // 6400
#define BDIM        256
#define NBLK        (NPAIR / BDIM)                // 25 blocks
#define W_BOX       5.0f
#define W_CLS       0.5f
#define W_DFL       1.5f

typedef float v2f __attribute__((ext_vector_type(2)));
typedef float v8f __attribute__((ext_vector_type(8)));

// -------------------------------------------------------------------
// Pass 1: one thread per (b,g) pair. Gather the selected cell's 84
// logits, compute box/cls/dfl losses, reduce within the block, write
// per-block partial sums (3 floats) to workspace. Wave32 shuffles.
// -------------------------------------------------------------------
__global__ __launch_bounds__(BDIM)
void yolo_partial_kernel(const float* __restrict__ pred0,
                         const float* __restrict__ pred1,
                         const float* __restrict__ pred2,
                         const float* __restrict__ boxes,
                         const int*   __restrict__ labels,
                         const int*   __restrict__ img_size_p,
                         float*       __restrict__ partials)
{
    const int t = blockIdx.x * BDIM + threadIdx.x;   // [0, 6400)
    const int b = t / NGT;

    const float img = (float)img_size_p[0];

    const float4 bx = ((const float4*)boxes)[t];
    const float cx = bx.x, cy = bx.y, w = bx.z, h = bx.w;
    const int lab = labels[t];

    // level select: scale = max(w,h)*img ; <64 -> 0, <128 -> 1, else 2
    const float scale = fmaxf(w, h) * img;
    const int lvl = (scale < 64.0f) ? 0 : ((scale < 128.0f) ? 1 : 2);
    const float* pred = (lvl == 0) ? pred0 : ((lvl == 1) ? pred1 : pred2);
    const int W = 80 >> lvl;            // 80 / 40 / 20
    const int H = W;
    const float Wf = (float)W, Hf = (float)H;

    int gx = (int)floorf(cx * Wf); gx = min(max(gx, 0), W - 1);
    int gy = (int)floorf(cy * Hf); gy = min(max(gy, 0), H - 1);

    const float* lg = pred + (size_t)(((b * H + gy) * W + gx)) * CPRED;

    const float ax = (float)gx + 0.5f;
    const float ay = (float)gy + 0.5f;
    const float hw = 0.5f * w, hh = 0.5f * h;
    const float x1 = (cx - hw) * Wf, x2 = (cx + hw) * Wf;
    const float y1 = (cy - hh) * Hf, y2 = (cy + hh) * Hf;

    float tt[4];
    tt[0] = fmaxf(ax - x1, 0.0f);
    tt[1] = fmaxf(ay - y1, 0.0f);
    tt[2] = fmaxf(x2 - ax, 0.0f);
    tt[3] = fmaxf(y2 - ay, 0.0f);

    // ---- DFL: per side, softmax over 16 bins (f32, log-sum-exp) ----
    float dfl = 0.0f;
    float dexp[4];
#pragma unroll
    for (int i = 0; i < 4; ++i) {
        float xv[REG_MAX];
        const float4* p4 = (const float4*)(lg + i * REG_MAX);  // 16B aligned (84*4=336)
#pragma unroll
        for (int q = 0; q < 4; ++q) {
            const float4 v = p4[q];
            xv[4 * q + 0] = v.x; xv[4 * q + 1] = v.y;
            xv[4 * q + 2] = v.z; xv[4 * q + 3] = v.w;
        }
        float m = xv[0];
#pragma unroll
        for (int j = 1; j < REG_MAX; ++j) m = fmaxf(m, xv[j]);
        float s = 0.0f, sw = 0.0f;
#pragma unroll
        for (int j = 0; j < REG_MAX; ++j) {
            const float e = expf(xv[j] - m);
            s  += e;
            sw += (float)j * e;
        }
        const float lse = m + logf(s);
        int tb = (int)tt[i];                 // tt >= 0, cast == floor
        tb = min(tb, REG_MAX - 1);
        dfl += (lse - xv[tb]);               // -log_softmax[tb]
        dexp[i] = sw / s;                    // sum_j j * softmax_j
    }
    dfl *= 0.25f;

    // ---- GIoU box loss ----
    const float x1p = (ax - dexp[0]) / Wf;
    const float y1p = (ay - dexp[1]) / Hf;
    const float x2p = (ax + dexp[2]) / Wf;
    const float y2p = (ay + dexp[3]) / Hf;
    const float gx1 = cx - hw, gy1 = cy - hh;
    const float gx2 = cx + hw, gy2 = cy + hh;

    const float iw = fmaxf(fminf(x2p, gx2) - fmaxf(x1p, gx1), 0.0f);
    const float ih = fmaxf(fminf(y2p, gy2) - fmaxf(y1p, gy1), 0.0f);
    const float inter  = iw * ih;
    const float area_p = (x2p - x1p) * (y2p - y1p);
    const float area_g = (gx2 - gx1) * (gy2 - gy1);
    const float uni    = area_p + area_g - inter;
    const float iou    = inter / uni;
    const float areac  = (fmaxf(x2p, gx2) - fminf(x1p, gx1)) *
                         (fmaxf(y2p, gy2) - fminf(y1p, gy1));
    const float giou   = iou - (areac - uni) / areac;
    const float box_loss = 1.0f - giou;

    // ---- BCE over 20 classes ----
    float bce = 0.0f;
    const float* cl = lg + 4 * REG_MAX;
#pragma unroll
    for (int c = 0; c < NUM_CLASSES; ++c) {
        const float z  = cl[c];
        const float on = (c == lab) ? 1.0f : 0.0f;
        bce += fmaxf(z, 0.0f) - z * on + log1pf(expf(-fabsf(z)));
    }
    const float cls_loss = bce * (1.0f / NUM_CLASSES);

    // ---- block reduction: wave32 shuffle then LDS across 8 waves ----
    float v0 = box_loss, v1 = cls_loss, v2 = dfl;
#pragma unroll
    for (int off = 16; off > 0; off >>= 1) {
        v0 += __shfl_down(v0, off, 32);
        v1 += __shfl_down(v1, off, 32);
        v2 += __shfl_down(v2, off, 32);
    }
    __shared__ float red[(BDIM / 32) * 3];
    const int wave = threadIdx.x >> 5;
    const int lane = threadIdx.x & 31;
    if (lane == 0) {
        red[wave * 3 + 0] = v0;
        red[wave * 3 + 1] = v1;
        red[wave * 3 + 2] = v2;
    }
    __syncthreads();
    if (threadIdx.x == 0) {
        float s0 = 0.0f, s1 = 0.0f, s2 = 0.0f;
#pragma unroll
        for (int i = 0; i < BDIM / 32; ++i) {
            s0 += red[i * 3 + 0];
            s1 += red[i * 3 + 1];
            s2 += red[i * 3 + 2];
        }
        partials[blockIdx.x * 3 + 0] = s0;
        partials[blockIdx.x * 3 + 1] = s1;
        partials[blockIdx.x * 3 + 2] = s2;
    }
}

// -------------------------------------------------------------------
// Pass 2: one wave32. Sum the 25 per-block partials per component with
// V_WMMA_F32_16X16X4_F32 (B = ones -> exact f32 row-sums on the matrix
// pipe), fold the 16x16 D tile with 7 VALU adds + one shfl_xor(16).
// A-layout (ISA 7.12.2, 32-bit A 16x4): lane<16 holds K=0,1 of row
// (lane&15); lane>=16 holds K=2,3. EXEC is all-ones (32 threads).
// -------------------------------------------------------------------
__global__ __launch_bounds__(32)
void yolo_final_kernel(const float* __restrict__ partials,
                       float*       __restrict__ out)
{
    const int lane = threadIdx.x;          // 0..31, all active
    const int row  = lane & 15;
    const int kbase = 4 * row + ((lane >> 4) << 1);   // K index of a.x

    v2f bones; bones.x = 1.0f; bones.y = 1.0f;        // B = all ones (layout-free)

    float sums[3];
#pragma unroll
    for (int c = 0; c < 3; ++c) {
        v2f a;
        a.x = (kbase     < NBLK) ? partials[(kbase)     * 3 + c] : 0.0f;
        a.y = (kbase + 1 < NBLK) ? partials[(kbase + 1) * 3 + c] : 0.0f;
        v8f acc = {};
        // D[m,n] = sum_k A[m,k] * 1 + 0  -> every column holds the row sums
        acc = __builtin_amdgcn_wmma_f32_16x16x4_f32(
                  /*neg_a=*/false, a, /*neg_b=*/false, bones,
                  /*c_mod=*/(short)0, acc, /*reuse_a=*/false, /*reuse_b=*/false);
        float s8 = acc[0] + acc[1] + acc[2] + acc[3] +
                   acc[4] + acc[5] + acc[6] + acc[7];     // rows M=0..7 (lane<16) / 8..15
        s8 += __shfl_xor(s8, 16, 32);                      // pair lanes l and l^16
        sums[c] = s8;                                      // total of all 64 (25 live) slots
    }

    if (lane == 0) {
        const float inv = 1.0f / (float)NPAIR;
        const float box  = sums[0] * inv;
        const float clsm = sums[1] * inv;
        const float dflm = sums[2] * inv;
        out[0] = W_BOX * box + W_CLS * clsm + W_DFL * dflm;
        out[1] = box;
        out[2] = clsm;
        out[3] = dflm;
    }
}

// -------------------------------------------------------------------
extern "C" void kernel_launch(void* const* d_in, const int* in_sizes, int n_in,
                              void* d_out, int out_size, void* d_ws, size_t ws_size,
                              hipStream_t stream)
{
    const float* pred0  = (const float*)d_in[0];
    const float* pred1  = (const float*)d_in[1];
    const float* pred2  = (const float*)d_in[2];
    const float* boxes  = (const float*)d_in[3];
    const int*   labels = (const int*)  d_in[4];
    const int*   imgsz  = (const int*)  d_in[5];

    float* partials = (float*)d_ws;          // NBLK * 3 floats, fully overwritten
    float* out      = (float*)d_out;

    yolo_partial_kernel<<<NBLK, BDIM, 0, stream>>>(pred0, pred1, pred2,
                                                   boxes, labels, imgsz, partials);
    yolo_final_kernel<<<1, 32, 0, stream>>>(partials, out);
}